// RNSCN_6330781795144
// MI455X (gfx1250) — compile-verified
//
#include <hip/hip_runtime.h>
#include <math.h>

typedef float v2f __attribute__((ext_vector_type(2)));
typedef float v8f __attribute__((ext_vector_type(8)));

#define B_   256
#define T_   128
#define DW_  300
#define H_   128
#define NT_  8
#define N_   (B_ * T_)
#define NBUCKET 96            // levels 1..12  x  8 types
#define ASTRIDE 308           // 4*odd -> conflict-free, 8B-aligned pairs
#define HSTRIDE 132           // 4*odd -> conflict-free, 8B-aligned pairs

// ---------------------------------------------------------------- init
__global__ void k_init(float* __restrict__ seg, int* __restrict__ cnt,
                       int* __restrict__ cur, int* __restrict__ bstart) {
    int gid = blockIdx.x * blockDim.x + threadIdx.x;
    if (gid < N_ * H_) seg[gid] = 0.0f;
    if (gid <= NBUCKET) bstart[gid] = 0;
    if (gid < NBUCKET) { cnt[gid] = 0; cur[gid] = 0; }
}

// ------------------------------------------------- parent idx + bucket count
__global__ void k_prep(const int* __restrict__ parent, const int* __restrict__ dep_type,
                       const int* __restrict__ height, int* __restrict__ parg,
                       int* __restrict__ cnt) {
    int i = blockIdx.x * blockDim.x + threadIdx.x;
    if (i >= N_) return;
    int par  = parent[i];
    int base = (i / T_) * T_;
    int pg   = (par >= 0) ? (base + par) : i;
    parg[i] = pg;
    if (par >= 0) {
        int lvl = height[pg];          // parent has a child -> lvl in 1..12
        int t   = dep_type[i];
        atomicAdd(&cnt[(lvl - 1) * NT_ + t], 1);
    }
}

// ---------------------------------------------------------------- scan
__global__ void k_scan(const int* __restrict__ cnt, int* __restrict__ bstart) {
    if (blockIdx.x == 0 && threadIdx.x == 0) {
        int acc = 0;
        for (int b = 0; b < NBUCKET; ++b) { bstart[b] = acc; acc += cnt[b]; }
        bstart[NBUCKET] = acc;
    }
}

// ---------------------------------------------------------------- fill
__global__ void k_fill(const int* __restrict__ parent, const int* __restrict__ dep_type,
                       const int* __restrict__ height, const int* __restrict__ parg,
                       const int* __restrict__ bstart, int* __restrict__ cur,
                       int* __restrict__ elist) {
    int i = blockIdx.x * blockDim.x + threadIdx.x;
    if (i >= N_) return;
    if (parent[i] < 0) return;
    int pg = parg[i];
    int bucket = (height[pg] - 1) * NT_ + dep_type[i];
    int pos = atomicAdd(&cur[bucket], 1);
    elist[bstart[bucket] + pos] = i;
}

// ------------------------------------------------- phase 1: wx = x @ W_wh^T
// block = 256 threads (8 waves); block computes a 16 x 128 tile of wx.
// Also writes h = tanh(wx + b)  (the "simple" value; final for leaves).
__global__ __launch_bounds__(256) void k_wx(const float* __restrict__ x,
                                            const float* __restrict__ Wwh,
                                            const float* __restrict__ bwh,
                                            float* __restrict__ wx,
                                            float* __restrict__ h) {
    __shared__ float At[16 * ASTRIDE];
    const int row0 = blockIdx.x * 16;
    const int tid  = threadIdx.x;

    for (int idx = tid; idx < 16 * DW_; idx += 256) {
        int r = idx / DW_;
        int k = idx - r * DW_;
        At[r * ASTRIDE + k] = x[(size_t)(row0 + r) * DW_ + k];
    }
    __syncthreads();

    const int wave = tid >> 5, lane = tid & 31;
    const int m = lane & 15, half = lane >> 4;
    const int n0 = wave * 16;
    const int ncol = n0 + m;

    v8f acc = {};
    const float* wrow = Wwh + (size_t)ncol * DW_;   // B[k][n] = W_wh[n][k]
    for (int s = 0; s < DW_ / 4; ++s) {
        int kA = 4 * s + 2 * half;
        v2f a, b;
        a[0] = At[m * ASTRIDE + kA];
        a[1] = At[m * ASTRIDE + kA + 1];
        b[0] = wrow[kA];
        b[1] = wrow[kA + 1];
        acc = __builtin_amdgcn_wmma_f32_16x16x4_f32(false, a, false, b,
                                                    (short)0, acc, false, false);
    }

    const float bb = bwh[ncol];
    for (int r = 0; r < 8; ++r) {
        int M = half ? (r + 8) : r;
        size_t o = (size_t)(row0 + M) * H_ + ncol;
        float pre = acc[r];
        wx[o] = pre;
        h[o]  = tanhf(pre + bb);                    // simple
    }
}

// ------------------------------------------------- phase 2: edge tiles
// grid = (2048, 8): x = 16-edge tile within bucket, y = relation type.
// Rel  = tanh( H16 @ W_hr^T + WXpar )   (16x128)
// Infl = Rel @ W_rel[t]^T               (16x128), scatter-added into seg[parent]
__global__ __launch_bounds__(256) void k_edge(int level,
        const float* __restrict__ Whr,  const float* __restrict__ Wrel,
        const float* __restrict__ wx,   const float* __restrict__ h,
        const int*  __restrict__ parg,  const int*  __restrict__ elist,
        const int*  __restrict__ bstart, const int* __restrict__ cnt,
        float* __restrict__ seg) {
    const int type   = blockIdx.y;
    const int bucket = (level - 1) * NT_ + type;
    const int count  = cnt[bucket];
    const int tile0  = blockIdx.x * 16;
    if (tile0 >= count) return;                     // block-uniform exit
    const int start  = bstart[bucket];

    __shared__ float Hl[16 * HSTRIDE];
    __shared__ float Wx[16 * HSTRIDE];
    __shared__ float Rl[16 * HSTRIDE];
    __shared__ int   eIdx[16];
    __shared__ int   pIdx[16];
    __shared__ int   vmask[16];

    const int tid = threadIdx.x;
    if (tid < 16) {
        int slot = tile0 + tid;
        int v = (slot < count) ? 1 : 0;
        int c = v ? elist[start + slot] : 0;
        int p = v ? parg[c] : 0;
        eIdx[tid] = c; pIdx[tid] = p; vmask[tid] = v;
    }
    __syncthreads();

    for (int idx = tid; idx < 16 * H_; idx += 256) {
        int i = idx >> 7, j = idx & 127;
        float hv = 0.0f, wv = 0.0f;
        if (vmask[i]) {
            hv = h [(size_t)eIdx[i] * H_ + j];      // child hidden (finalized)
            wv = wx[(size_t)pIdx[i] * H_ + j];      // parent focus role
        }
        Hl[i * HSTRIDE + j] = hv;
        Wx[i * HSTRIDE + j] = wv;
    }
    __syncthreads();

    const int wave = tid >> 5, lane = tid & 31;
    const int m = lane & 15, half = lane >> 4;
    const int n0 = wave * 16;
    const int ncol = n0 + m;

    // ---- GEMM 1: Rel = tanh(Hl @ Whr^T + Wx)
    v8f acc = {};
    const float* wrow = Whr + (size_t)ncol * H_;    // B[k][n] = W_hr[n][k]
    for (int s = 0; s < H_ / 4; ++s) {
        int kA = 4 * s + 2 * half;
        v2f a, b;
        a[0] = Hl[m * HSTRIDE + kA];
        a[1] = Hl[m * HSTRIDE + kA + 1];
        b[0] = wrow[kA];
        b[1] = wrow[kA + 1];
        acc = __builtin_amdgcn_wmma_f32_16x16x4_f32(false, a, false, b,
                                                    (short)0, acc, false, false);
    }
    for (int r = 0; r < 8; ++r) {
        int M = half ? (r + 8) : r;
        Rl[M * HSTRIDE + ncol] = tanhf(acc[r] + Wx[M * HSTRIDE + ncol]);
    }
    __syncthreads();

    // ---- GEMM 2: Infl = Rl @ Wrel[type]^T, scatter into seg[parent]
    v8f acc2 = {};
    const float* wrow2 = Wrel + ((size_t)type * H_ + ncol) * H_;
    for (int s = 0; s < H_ / 4; ++s) {
        int kA = 4 * s + 2 * half;
        v2f a, b;
        a[0] = Rl[m * HSTRIDE + kA];
        a[1] = Rl[m * HSTRIDE + kA + 1];
        b[0] = wrow2[kA];
        b[1] = wrow2[kA + 1];
        acc2 = __builtin_amdgcn_wmma_f32_16x16x4_f32(false, a, false, b,
                                                     (short)0, acc2, false, false);
    }
    for (int r = 0; r < 8; ++r) {
        int M = half ? (r + 8) : r;
        if (vmask[M])
            atomicAdd(&seg[(size_t)pIdx[M] * H_ + ncol], acc2[r]);
    }
}

// ------------------------------------------------- finalize nodes at `level`
__global__ void k_final(int level, const int* __restrict__ height,
                        const float* __restrict__ seg, const float* __restrict__ wx,
                        const float* __restrict__ bwh, float* __restrict__ h) {
    int gid = blockIdx.x * blockDim.x + threadIdx.x;
    if (gid >= N_ * H_) return;
    int i = gid >> 7, j = gid & 127;
    if (height[i] != level) return;
    float simple = tanhf(wx[gid] + bwh[j]);
    h[gid] = tanhf(seg[gid] + simple);
}

// ----------------------------------------------------------------- launch
extern "C" void kernel_launch(void* const* d_in, const int* in_sizes, int n_in,
                              void* d_out, int out_size, void* d_ws, size_t ws_size,
                              hipStream_t stream) {
    const float* x       = (const float*)d_in[0];   // [B,T,DW]
    const float* Wwh     = (const float*)d_in[1];   // [H,DW]
    const float* bwh     = (const float*)d_in[2];   // [H]
    const float* Whr     = (const float*)d_in[3];   // [H,H]
    const float* Wrel    = (const float*)d_in[4];   // [NTYPES,H,H]
    const int*   parent  = (const int*)d_in[5];     // [B,T]
    const int*   dep_type= (const int*)d_in[6];     // [B,T]
    const int*   height  = (const int*)d_in[7];     // [B,T]
    float* h = (float*)d_out;                       // [B,T,H] — doubles as hidden state

    char* p = (char*)d_ws;
    float* wx   = (float*)p;  p += (size_t)N_ * H_ * sizeof(float);
    float* seg  = (float*)p;  p += (size_t)N_ * H_ * sizeof(float);
    int*   parg = (int*)p;    p += (size_t)N_ * sizeof(int);
    int*   elist= (int*)p;    p += (size_t)N_ * sizeof(int);
    int*   cnt  = (int*)p;    p += 128 * sizeof(int);
    int*   cur  = (int*)p;    p += 128 * sizeof(int);
    int*   bstart=(int*)p;    p += 128 * sizeof(int);

    const int elemGrid = (N_ * H_ + 255) / 256;     // 16384
    const int nodeGrid = (N_ + 255) / 256;          // 128

    k_init<<<elemGrid, 256, 0, stream>>>(seg, cnt, cur, bstart);
    k_prep<<<nodeGrid, 256, 0, stream>>>(parent, dep_type, height, parg, cnt);
    k_scan<<<1, 32, 0, stream>>>(cnt, bstart);
    k_fill<<<nodeGrid, 256, 0, stream>>>(parent, dep_type, height, parg, bstart, cur, elist);
    k_wx<<<N_ / 16, 256, 0, stream>>>(x, Wwh, bwh, wx, h);

    for (int lvl = 1; lvl <= 12; ++lvl) {
        k_edge<<<dim3(N_ / 16, NT_), 256, 0, stream>>>(lvl, Whr, Wrel, wx, h,
                                                       parg, elist, bstart, cnt, seg);
        k_final<<<elemGrid, 256, 0, stream>>>(lvl, height, seg, wx, bwh, h);
    }
}